// WaveConv1d_73967926771991
// MI455X (gfx1250) — compile-verified
//
#include <hip/hip_runtime.h>
#include <hip/hip_bf16.h>

typedef __attribute__((ext_vector_type(2))) float v2f;
typedef __attribute__((ext_vector_type(8))) float v8f;

// ---- wavelet filters (db6-like, from reference DEC_LO) ----
// Analysis (correlation form, stride 2):  HLO[t] = DEC_LO[11-t], HHI[t] = (-1)^t * DEC_LO[t]
__constant__ float c_hlo[12] = {
     0.11154074335008017f,  0.4946238903983854f,   0.7511339080215775f,
     0.3152503517092432f,  -0.22626469396516913f, -0.12976686756709563f,
     0.09750160558707936f,  0.02752286553001629f, -0.031582039318031156f,
     0.0005538422009938016f, 0.004777257511010651f, -0.00107730108499558f };
__constant__ float c_hhi[12] = {
    -0.00107730108499558f, -0.004777257511010651f, 0.0005538422009938016f,
     0.031582039318031156f, 0.02752286553001629f, -0.09750160558707936f,
    -0.12976686756709563f,  0.22626469396516913f,  0.3152503517092432f,
    -0.7511339080215775f,   0.4946238903983854f,  -0.11154074335008017f };
// Synthesis (correlation on 2x-dilated, pad 1): GLO[t]=DEC_LO[t], GHI[t]=(-1)^(t+1)*DEC_LO[11-t]
__constant__ float c_glo[12] = {
    -0.00107730108499558f,  0.004777257511010651f, 0.0005538422009938016f,
    -0.031582039318031156f, 0.02752286553001629f,  0.09750160558707936f,
    -0.12976686756709563f, -0.22626469396516913f,  0.3152503517092432f,
     0.7511339080215775f,   0.4946238903983854f,   0.11154074335008017f };
__constant__ float c_ghi[12] = {
    -0.11154074335008017f,  0.4946238903983854f,  -0.7511339080215775f,
     0.3152503517092432f,   0.22626469396516913f, -0.12976686756709563f,
    -0.09750160558707936f,  0.02752286553001629f,  0.031582039318031156f,
     0.0005538422009938016f, -0.004777257511010651f, -0.00107730108499558f };

// One DWT level: rows of length n -> lo/hi rows of length outsize.
// y[k] = sum_t xp[2k+t]*h[t], xp = symmetric-pad(x, padL left).
__global__ __launch_bounds__(256) void dwt_level(
    const float* __restrict__ in, long long inStride, int n,
    float* __restrict__ lo, float* __restrict__ hi,
    long long outStride, int outsize, int padL)
{
    __shared__ float s[522];           // window for 256 outputs: 2*255 + 12
    const int row = blockIdx.y;
    const int k0  = blockIdx.x * 256;
    const float* src = in + (size_t)row * inStride;
    const int base = 2 * k0 - padL;    // input index of s[0]
    for (int o = threadIdx.x; o < 522; o += 256) {
        int idx = base + o;
        if (idx < 0)  idx = -idx - 1;          // symmetric reflect (half-sample)
        if (idx >= n) idx = 2 * n - 1 - idx;
        if (idx < 0)  idx = 0;                 // clamp: only hit by unused tail
        if (idx >= n) idx = n - 1;
        s[o] = src[idx];
    }
    __syncthreads();
    const int k = k0 + (int)threadIdx.x;
    if (k < outsize) {
        float aLo = 0.f, aHi = 0.f;
        const int off = 2 * (int)threadIdx.x;
        #pragma unroll
        for (int t = 0; t < 12; ++t) {
            const float v = s[off + t];
            aLo += v * c_hlo[t];
            aHi += v * c_hhi[t];
        }
        lo[(size_t)row * outStride + k] = aLo;
        hi[(size_t)row * outStride + k] = aHi;
    }
}

// One IDWT level: y[m] = sum_{t:(m+t-1) even, 0<=j<inLen} lo[j]*GLO[t]+hi[j]*GHI[t], j=(m+t-1)/2
__global__ __launch_bounds__(256) void idwt_level(
    const float* __restrict__ lo, long long loStride,
    const float* __restrict__ hi, long long hiStride,
    float* __restrict__ out, long long outStride,
    int inLen, int outLen)
{
    __shared__ float slo[136];
    __shared__ float shi[136];
    const int row = blockIdx.y;
    const int m0  = blockIdx.x * 256;
    const int jBase = (m0 > 0) ? ((m0 - 1) >> 1) : 0;
    const int jEnd  = (m0 + 265) >> 1;
    const int cnt   = jEnd - jBase + 1;        // <= 134
    const float* lsrc = lo + (size_t)row * loStride;
    const float* hsrc = hi + (size_t)row * hiStride;
    for (int o = threadIdx.x; o < cnt; o += 256) {
        const int j = jBase + o;
        const bool ok = (j < inLen);
        slo[o] = ok ? lsrc[j] : 0.0f;
        shi[o] = ok ? hsrc[j] : 0.0f;
    }
    __syncthreads();
    const int m = m0 + (int)threadIdx.x;
    if (m < outLen) {
        float acc = 0.f;
        const int t0 = (m + 1) & 1;            // parity making m+t-1 even
        #pragma unroll
        for (int sidx = 0; sidx < 6; ++sidx) {
            const int t = t0 + 2 * sidx;
            const int d = m + t - 1;
            if (d >= 0) {
                const int o = (d >> 1) - jBase;
                acc += slo[o] * c_glo[t] + shi[o] * c_ghi[t];
            }
        }
        out[(size_t)row * outStride + m] = acc;
    }
}

// Per-mode GEMM: out[b, o, pos] = sum_i A[b, i, pos] * W[i, o, pos]
// One wave computes the full 16(batch) x 16(Co-tile) output with K=256 via
// V_WMMA_F32_16X16X4_F32 (fp32 in/out -> no precision loss; compute is not the
// bottleneck at 2.2 GFLOP total).
// A-layout (ISA 7.12.2, 32-bit A 16x4): lane<16: M=lane, VGPRs hold K={0,1};
// lane>=16: M=lane-16, K={2,3}. B mirrored with N=lane. C/D: VGPR j -> M=j+8*half.
__global__ __launch_bounds__(32) void modal_gemm(
    const float* __restrict__ A, long long aStride,   // row (b*256+i), col pos
    const float* __restrict__ W,                      // (i*256+o)*modes + pos
    float* __restrict__ Y, long long yStride,         // row (b*256+o), col pos
    int modes)
{
    const int pos   = blockIdx.x;
    const int oBase = blockIdx.y * 16;
    const int lane  = threadIdx.x;
    const int half  = lane >> 4;
    const int col   = lane & 15;
    v8f c = {};
    #pragma unroll 4
    for (int k0 = 0; k0 < 256; k0 += 4) {
        const int ka = k0 + 2 * half;
        v2f a, b;
        a.x = A[(size_t)(col * 256 + ka)     * (size_t)aStride + pos];
        a.y = A[(size_t)(col * 256 + ka + 1) * (size_t)aStride + pos];
        b.x = W[((size_t)ka       * 256 + oBase + col) * (size_t)modes + pos];
        b.y = W[((size_t)(ka + 1) * 256 + oBase + col) * (size_t)modes + pos];
        c = __builtin_amdgcn_wmma_f32_16x16x4_f32(
                false, a, false, b, (short)0, c, false, false);
    }
    #pragma unroll
    for (int j = 0; j < 8; ++j) {
        const int m = j + 8 * half;            // batch index
        Y[(size_t)(m * 256 + oBase + col) * (size_t)yStride + pos] = c[j];
    }
}

extern "C" void kernel_launch(void* const* d_in, const int* in_sizes, int n_in,
                              void* d_out, int out_size, void* d_ws, size_t ws_size,
                              hipStream_t stream) {
    (void)in_sizes; (void)n_in; (void)out_size; (void)ws_size;
    const float* x  = (const float*)d_in[0];   // (16, 256, 8192)
    const float* w1 = (const float*)d_in[1];   // (256, 256, 522)
    const float* w2 = (const float*)d_in[2];   // (256, 256, 522)
    float* out = (float*)d_out;                // (16, 256, 8192)

    const long long R = 4096;                  // B*C rows
    const int N0 = 8192, N1 = 4101, N2 = 2056, N3 = 1033, N4 = 522;
    const int R2LEN = 4102;                    // idwt level-2 output length

    float* ws = (float*)d_ws;
    size_t off = 0;
    float* hi1 = ws + off; off += (size_t)R * N1;
    float* hi2 = ws + off; off += (size_t)R * N2;
    float* hi3 = ws + off; off += (size_t)R * N3;
    float* hi4 = ws + off; off += (size_t)R * N4;
    float* lo1 = ws + off; off += (size_t)R * N1;
    float* lo2 = ws + off; off += (size_t)R * N2;
    float* lo3 = ws + off; off += (size_t)R * N3;
    float* lo4 = ws + off; off += (size_t)R * N4;
    float* gHi = ws + off; off += (size_t)R * N4;
    float* r2o = ws + off; off += (size_t)R * R2LEN;
    // region reuse (consumers of the original contents have already run):
    float* gLo = lo3;   // stride N3, uses first 522 of each row
    float* r4o = lo2;   // stride N2, uses first 1034
    float* r3o = lo1;   // stride N1, uses first 2056

    dim3 blk(256);
    // ---- analysis DWT (4 levels), padL = floor(p/2) = 10 at every level ----
    dwt_level<<<dim3((N1 + 255) / 256, R), blk, 0, stream>>>(x,   N0, N0, lo1, hi1, N1, N1, 10);
    dwt_level<<<dim3((N2 + 255) / 256, R), blk, 0, stream>>>(lo1, N1, N1, lo2, hi2, N2, N2, 10);
    dwt_level<<<dim3((N3 + 255) / 256, R), blk, 0, stream>>>(lo2, N2, N2, lo3, hi3, N3, N3, 10);
    dwt_level<<<dim3((N4 + 255) / 256, R), blk, 0, stream>>>(lo3, N3, N3, lo4, hi4, N4, N4, 10);

    // ---- per-mode channel-mixing GEMMs (WMMA fp32) ----
    modal_gemm<<<dim3(N4, 16), dim3(32), 0, stream>>>(lo4, N4, w1, gLo, N3, N4);
    modal_gemm<<<dim3(N4, 16), dim3(32), 0, stream>>>(hi4, N4, w2, gHi, N4, N4);

    // ---- synthesis IDWT (4 levels, x0 truncated to hi length via inLen) ----
    idwt_level<<<dim3((1034 + 255) / 256, R), blk, 0, stream>>>(gLo, N3, gHi, N4, r4o, N2, N4, 1034);
    idwt_level<<<dim3((N2 + 255) / 256,  R), blk, 0, stream>>>(r4o, N2, hi3, N3, r3o, N1, N3, N2);
    idwt_level<<<dim3((R2LEN + 255) / 256, R), blk, 0, stream>>>(r3o, N1, hi2, N2, r2o, R2LEN, N2, R2LEN);
    idwt_level<<<dim3((N0 + 255) / 256,  R), blk, 0, stream>>>(r2o, R2LEN, hi1, N1, out, N0, N1, N0);
}